// ResNet_16690242912796
// MI455X (gfx1250) — compile-verified
//
#include <hip/hip_runtime.h>

#define C_DIM     128
#define W_STRIDE  132   // padded row stride (floats) for transposed W in LDS
#define X_STRIDE  132   // padded row stride (floats) for gathered x tile
#define NWAVES    4
#define TILE_M    32    // pairs per wave (two 16-row WMMA tiles share each B fragment)

typedef __attribute__((ext_vector_type(2))) float v2f;
typedef __attribute__((ext_vector_type(8))) float v8f;

// ---------------------------------------------------------------------------
// Sparse conv layer: y[out_map[k,p]] += x[in_map[k,p]] @ W[k]
// One wave handles a 32-pair micro-tile; fp32 WMMA 16x16x4.
// Each B fragment (W^T from LDS) feeds two WMMAs (M=0..15 and M=16..31).
// The N dimension (8 tiles of 16 cols) is processed in two passes of 4 tiles
// to cap live accumulators at 8 x v8f (64 VGPRs) and avoid spills.
// ---------------------------------------------------------------------------
__global__ __launch_bounds__(128) void sparse_conv_wmma(
    const float* __restrict__ x,       // [N,128]
    const float* __restrict__ W,       // [K,128,128]
    const int*   __restrict__ in_map,  // [K,P]
    const int*   __restrict__ out_map, // [K,P]
    float*       __restrict__ y,       // [N,128] pre-zeroed accumulator
    int P)
{
  __shared__ __align__(16) float lds_w[C_DIM * W_STRIDE];          // W[k]^T : [col][row]
  __shared__ __align__(16) float lds_x[NWAVES * TILE_M * X_STRIDE];

  const int k    = blockIdx.y;
  const int tid  = threadIdx.x;
  const int lane = tid & 31;
  const int wave = tid >> 5;

  // ---- stage W[k] transposed into LDS (coalesced global reads) ----
  const float* Wk = W + (size_t)k * C_DIM * C_DIM;
  for (int idx = tid; idx < C_DIM * (C_DIM / 4); idx += 128) {
    int row = idx >> 5;             // 0..127
    int c4  = (idx & 31) << 2;      // 0,4,...,124
    float4 v = ((const float4*)(Wk + (size_t)row * C_DIM))[idx & 31];
    lds_w[(c4 + 0) * W_STRIDE + row] = v.x;
    lds_w[(c4 + 1) * W_STRIDE + row] = v.y;
    lds_w[(c4 + 2) * W_STRIDE + row] = v.z;
    lds_w[(c4 + 3) * W_STRIDE + row] = v.w;
  }
  __syncthreads();

  const int base = (blockIdx.x * NWAVES + wave) * TILE_M;
  if (base >= P) return;            // wave-uniform exit (terminated waves release barrier)

  const int m_lane = lane & 15;
  int pidx   = base + lane;         // this wave covers pairs base..base+31
  int pcl    = pidx < P ? pidx : P - 1;
  int rin    = in_map [(size_t)k * P + pcl];
  int rout   = out_map[(size_t)k * P + pcl];
  int lvalid = pidx < P ? 1 : 0;

  // ---- gather 32 rows of x into this wave's LDS tile ----
  float* xw = lds_x + wave * TILE_M * X_STRIDE;
  for (int i = 0; i < TILE_M; ++i) {
    int row = __shfl(rin, i);
    int v   = __shfl(lvalid, i);
    float4 val = make_float4(0.f, 0.f, 0.f, 0.f);
    if (v) val = ((const float4*)(x + (size_t)row * C_DIM))[lane];
    ((float4*)(xw + i * X_STRIDE))[lane] = val;  // 512B coalesced per row
  }
  // LDS is in-order within a wave; fence the compiler + hw counter anyway.
  asm volatile("s_wait_dscnt 0" ::: "memory");

  const int khalf = (lane >> 4) << 1;   // lanes 0-15 -> K+{0,1}; 16-31 -> K+{2,3}
  const int mbase = (lane >> 4) << 3;   // D layout: M = r + 8*(lane>=16)

  #pragma unroll 1
  for (int half = 0; half < 2; ++half) {
    const int n0 = half * 4;            // n-tiles n0..n0+3 this pass

    v8f acc0[4], acc1[4];
    #pragma unroll
    for (int n = 0; n < 4; ++n) {
      acc0[n] = (v8f){0.f,0.f,0.f,0.f,0.f,0.f,0.f,0.f};
      acc1[n] = (v8f){0.f,0.f,0.f,0.f,0.f,0.f,0.f,0.f};
    }

    // ---- [32x128] @ [128x64] via 256 x v_wmma_f32_16x16x4_f32 per pass ----
    for (int ks = 0; ks < 32; ++ks) {
      int kb = ks << 2;
      v2f a0 = *(const v2f*)(xw + m_lane * X_STRIDE + kb + khalf);
      v2f a1 = *(const v2f*)(xw + (16 + m_lane) * X_STRIDE + kb + khalf);
      #pragma unroll
      for (int n = 0; n < 4; ++n) {
        v2f b = *(const v2f*)(lds_w + ((n0 + n) * 16 + m_lane) * W_STRIDE + kb + khalf);
        acc0[n] = __builtin_amdgcn_wmma_f32_16x16x4_f32(
                      false, a0, false, b, (short)0, acc0[n], false, false);
        acc1[n] = __builtin_amdgcn_wmma_f32_16x16x4_f32(
                      false, a1, false, b, (short)0, acc1[n], false, false);
      }
    }

    // ---- scatter-add: N = (n0+n)*16 + (lane&15) ----
    #pragma unroll
    for (int r = 0; r < 8; ++r) {
      int m = mbase + r;
      // tile 0: pairs base+0..15
      int row0 = __shfl(rout, m);
      int v0   = __shfl(lvalid, m);
      if (v0) {
        float* yrow = y + (size_t)row0 * C_DIM + n0 * 16 + m_lane;
        #pragma unroll
        for (int n = 0; n < 4; ++n)
          atomicAdd(yrow + n * 16, acc0[n][r]);
      }
      // tile 1: pairs base+16..31
      int row1 = __shfl(rout, 16 + m);
      int v1   = __shfl(lvalid, 16 + m);
      if (v1) {
        float* yrow = y + (size_t)row1 * C_DIM + n0 * 16 + m_lane;
        #pragma unroll
        for (int n = 0; n < 4; ++n)
          atomicAdd(yrow + n * 16, acc1[n][r]);
      }
    }
  }
}

// ---------------------------------------------------------------------------
// Elementwise helpers (float4-vectorized; C=128 -> 32 float4 per row)
// ---------------------------------------------------------------------------
__global__ void zero2_kernel(float4* a, float4* b, size_t n4) {
  size_t i = (size_t)blockIdx.x * blockDim.x + threadIdx.x;
  if (i < n4) {
    float4 z = make_float4(0.f, 0.f, 0.f, 0.f);
    a[i] = z; b[i] = z;
  }
}

__global__ void bias_relu_kernel(float* y, const float* bias, size_t n4) {
  size_t i = (size_t)blockIdx.x * blockDim.x + threadIdx.x;
  if (i >= n4) return;
  float4 v = ((float4*)y)[i];
  float4 b = ((const float4*)bias)[i & 31];
  v.x = fmaxf(v.x + b.x, 0.f);
  v.y = fmaxf(v.y + b.y, 0.f);
  v.z = fmaxf(v.z + b.z, 0.f);
  v.w = fmaxf(v.w + b.w, 0.f);
  ((float4*)y)[i] = v;
}

__global__ void final_kernel(float* out, const float* bias,
                             const float* feat, size_t n4) {
  size_t i = (size_t)blockIdx.x * blockDim.x + threadIdx.x;
  if (i >= n4) return;
  float4 o = ((float4*)out)[i];
  float4 b = ((const float4*)bias)[i & 31];
  float4 f = ((const float4*)feat)[i];
  o.x += b.x + f.x; o.y += b.y + f.y;
  o.z += b.z + f.z; o.w += b.w + f.w;
  ((float4*)out)[i] = o;
}

// ---------------------------------------------------------------------------
extern "C" void kernel_launch(void* const* d_in, const int* in_sizes, int n_in,
                              void* d_out, int out_size, void* d_ws, size_t ws_size,
                              hipStream_t stream) {
  const float* features = (const float*)d_in[0];
  const int*   in_map   = (const int*)  d_in[1];
  const int*   out_map  = (const int*)  d_in[2];
  const float* W0       = (const float*)d_in[3];
  const float* b0       = (const float*)d_in[4];
  const float* W1       = (const float*)d_in[5];
  const float* b1       = (const float*)d_in[6];
  float*       out      = (float*)d_out;

  const int N = in_sizes[0] / C_DIM;
  const int K = in_sizes[3] / (C_DIM * C_DIM);   // 27
  const int P = in_sizes[1] / K;                 // 50000

  float* y0 = (float*)d_ws;                      // [N,128] intermediate

  const size_t total = (size_t)N * C_DIM;
  const size_t n4    = total / 4;
  const int    eb    = 256;
  const int    eg    = (int)((n4 + eb - 1) / eb);

  dim3 cgrid((P + NWAVES * TILE_M - 1) / (NWAVES * TILE_M), K);

  zero2_kernel   <<<eg, eb, 0, stream>>>((float4*)y0, (float4*)out, n4);
  sparse_conv_wmma<<<cgrid, 128, 0, stream>>>(features, W0, in_map, out_map, y0, P);
  bias_relu_kernel<<<eg, eb, 0, stream>>>(y0, b0, n4);
  sparse_conv_wmma<<<cgrid, 128, 0, stream>>>(y0, W1, in_map, out_map, out, P);
  final_kernel   <<<eg, eb, 0, stream>>>(out, b1, features, n4);
}